// GCN_86045374808289
// MI455X (gfx1250) — compile-verified
//
#include <hip/hip_runtime.h>

typedef __attribute__((ext_vector_type(2))) float v2f;
typedef __attribute__((ext_vector_type(8))) float v8f;
typedef __attribute__((ext_vector_type(4))) unsigned int v4u;
typedef __attribute__((ext_vector_type(8))) int v8i;
typedef __attribute__((ext_vector_type(4))) int v4i;

// ---------------------------------------------------------------------------
// Degree / normalization kernels
// ---------------------------------------------------------------------------
__global__ void __launch_bounds__(256)
gcn_init_dinv(float* __restrict__ dinv, int n) {
    int i = blockIdx.x * blockDim.x + threadIdx.x;
    if (i < n) dinv[i] = 1.0f;  // self-loop contributes 1 to degree
}

__global__ void __launch_bounds__(256)
gcn_degree(const int* __restrict__ dst, float* __restrict__ dinv, int E) {
    int e = blockIdx.x * blockDim.x + threadIdx.x;
    if (e < E) {
        __hip_atomic_fetch_add(&dinv[dst[e]], 1.0f,
                               __ATOMIC_RELAXED, __HIP_MEMORY_SCOPE_AGENT);
    }
}

__global__ void __launch_bounds__(256)
gcn_rsqrt(float* __restrict__ dinv, int n) {
    int i = blockIdx.x * blockDim.x + threadIdx.x;
    if (i < n) dinv[i] = rsqrtf(dinv[i]);  // deg >= 1 always
}

// ---------------------------------------------------------------------------
// FP32 WMMA GEMM:  C[n_rows x NCOLS] = A[n_rows x 128] * W[128 x NCOLS]
//   SCALE: pre-scale A rows by dinv[row] (== post-scaling the output rows)
//   BIAS : add bias[col] in the epilogue
// A tile (16x128) staged via VGPRs (needs dinv scaling); the whole W
// (128xNCOLS) is DMA'd into LDS by the Tensor Data Mover, with D# padding
// reproducing the bank-conflict-free stride LDW = NCOLS+16.
// One 16x16 tile per wave, 8 waves/block. K fixed at 128.
// ---------------------------------------------------------------------------
template <int NCOLS, bool SCALE, bool BIAS>
__global__ void __launch_bounds__(256)
gcn_gemm_wmma_f32(const float* __restrict__ A,
                  const float* __restrict__ W,
                  const float* __restrict__ dinv,
                  const float* __restrict__ bias,
                  float* __restrict__ C,
                  int n_rows)
{
    constexpr int K   = 128;
    constexpr int LDA = 132;         // A tile stride: 16-lane reads hit 16 banks
    constexpr int LDW = NCOLS + 16;  // W stride: rows k,k+2 land 32 banks apart
    // pad after every NCOLS dwords: interval code i -> 2^(i+1) dwords
    constexpr int PAD_IVL = (NCOLS == 128) ? 6 : 5;
    constexpr int PAD_AMT = 15;      // code 15 -> 16 dwords = LDW - NCOLS
    __shared__ float As[16 * LDA];
    __shared__ float Ws[K * LDW];

    const int tid  = threadIdx.x;
    const int lane = tid & 31;
    const int wave = tid >> 5;
    const int m0   = blockIdx.x << 4;

    // ---- wave 0: kick off TDM load of W (128 x NCOLS) into padded LDS ----
    if (wave == 0) {
        unsigned lds_off = (unsigned)(unsigned long long)(uintptr_t)&Ws[0];
        unsigned long long ga = (unsigned long long)(uintptr_t)W;
        v4u g0;
        g0[0] = 1u;                                  // count=1 (valid user D#)
        g0[1] = lds_off;                             // lds_addr (bytes)
        g0[2] = (unsigned)(ga & 0xffffffffu);        // global_addr[31:0]
        g0[3] = (unsigned)((ga >> 32) & 0x01ffffffu) // global_addr[56:32]
              | (2u << 30);                          // type = 2 ("image")
        v8i g1;
        g1[0] = (2 << 16)                            // data_size = 4 bytes
              | (1 << 20)                            // pad_enable
              | (PAD_IVL << 22)                      // pad every NCOLS dwords
              | (PAD_AMT << 25);                     // pad 16 dwords
        g1[1] = (NCOLS & 0xffff) << 16;              // tensor_dim0[15:0]
        g1[2] = ((unsigned)NCOLS >> 16)              // tensor_dim0[31:16]
              | ((K & 0xffff) << 16);                // tensor_dim1[15:0]
        g1[3] = ((unsigned)K >> 16)                  // tensor_dim1[31:16]
              | (NCOLS << 16);                       // tile_dim0 = NCOLS
        g1[4] = K & 0xffff;                          // tile_dim1 = K rows
        g1[5] = NCOLS;                               // tensor_dim0_stride[31:0]
        g1[6] = 0;                                   // stride0 hi / stride1 lo
        g1[7] = 0;
        v4i gz4 = {0, 0, 0, 0};                      // 2D tile: groups 2/3 unused
        v8i gz8 = {0, 0, 0, 0, 0, 0, 0, 0};
        __builtin_amdgcn_tensor_load_to_lds(g0, g1, gz4, gz4, gz8, 0);
    }

    // ---- stage A tile (16 x 128), optionally scaled by dinv[row] ----
    {
        int r = tid >> 4;                  // 0..15
        int c = (tid & 15) << 3;           // 0,8,...,120
        int row = m0 + r;
        if (row >= n_rows) row = n_rows - 1;
        const float4* src = (const float4*)(A + (size_t)row * K + c);
        float4 v0 = src[0];
        float4 v1 = src[1];
        if (SCALE) {
            float di = dinv[row];
            v0.x *= di; v0.y *= di; v0.z *= di; v0.w *= di;
            v1.x *= di; v1.y *= di; v1.z *= di; v1.w *= di;
        }
        *(float4*)(As + r * LDA + c)     = v0;
        *(float4*)(As + r * LDA + c + 4) = v1;
    }

    if (wave == 0) {
        __builtin_amdgcn_s_wait_tensorcnt(0);        // TDM done before barrier
    }
    __syncthreads();

    const int ntile = wave;
    if ((ntile << 4) < NCOLS) {            // all 8 waves for 128, 4 for 64
        const int ml   = lane & 15;        // A row / B,C col within tile
        const int koff = (lane >> 4) << 1; // 0 (lanes 0-15) or 2 (lanes 16-31)
        const int n0   = ntile << 4;

        const float* Ap = As + ml * LDA + koff;
        const float* Bp = Ws + koff * LDW + n0 + ml;

        v8f acc = {};
#pragma unroll 8
        for (int k0 = 0; k0 < K; k0 += 4) {
            v2f a = *(const v2f*)(Ap + k0);      // A[m][k0+koff .. +1]
            v2f b;
            b.x = Bp[k0 * LDW];                  // W[k0+koff  ][n]
            b.y = Bp[(k0 + 1) * LDW];            // W[k0+koff+1][n]
            acc = __builtin_amdgcn_wmma_f32_16x16x4_f32(
                false, a, false, b, (short)0, acc, false, false);
        }

        // C/D layout: VGPR r -> M = r + (lane>=16 ? 8 : 0), N = lane%16
        const int rbase = (lane >> 4) << 3;      // 0 or 8
        const int col   = n0 + ml;
        const float bval = BIAS ? bias[col] : 0.0f;
        float* Cp = C + (size_t)(m0 + rbase) * NCOLS + col;

        if (m0 + 16 <= n_rows) {                 // full tile: unguarded stores
#pragma unroll
            for (int r = 0; r < 8; ++r) {
                float v = acc[r];
                if (BIAS) v += bval;             // compile-time dead when !BIAS
                Cp[r * NCOLS] = v;
            }
        } else {                                 // ragged edge
#pragma unroll
            for (int r = 0; r < 8; ++r) {
                if (m0 + rbase + r < n_rows) {
                    float v = acc[r];
                    if (BIAS) v += bval;
                    Cp[r * NCOLS] = v;
                }
            }
        }
    }
}

// ---------------------------------------------------------------------------
// Edge scatter: acc[dst] += g[src]; one edge per wave, float4 per lane.
// ---------------------------------------------------------------------------
__global__ void __launch_bounds__(256)
gcn_scatter(const int* __restrict__ src, const int* __restrict__ dst,
            const float* __restrict__ g, float* __restrict__ acc, int E)
{
    long long idx = (long long)blockIdx.x * blockDim.x + threadIdx.x;
    int e = (int)(idx >> 5);
    if (e >= E) return;
    int f = ((int)idx & 31) << 2;              // feature offset 0..124
    int s = src[e];
    int d = dst[e];
    float4 gv = *(const float4*)(g + (size_t)s * 128 + f);
    float* ap = acc + (size_t)d * 128 + f;
    __hip_atomic_fetch_add(ap + 0, gv.x, __ATOMIC_RELAXED, __HIP_MEMORY_SCOPE_AGENT);
    __hip_atomic_fetch_add(ap + 1, gv.y, __ATOMIC_RELAXED, __HIP_MEMORY_SCOPE_AGENT);
    __hip_atomic_fetch_add(ap + 2, gv.z, __ATOMIC_RELAXED, __HIP_MEMORY_SCOPE_AGENT);
    __hip_atomic_fetch_add(ap + 3, gv.w, __ATOMIC_RELAXED, __HIP_MEMORY_SCOPE_AGENT);
}

// ---------------------------------------------------------------------------
// Combine: acc[i] = relu(dinv[node]*(acc[i] + g[i]) + bias[f])  (in place)
// ---------------------------------------------------------------------------
__global__ void __launch_bounds__(256)
gcn_combine(const float* __restrict__ g, float* __restrict__ acc,
            const float* __restrict__ dinv, const float* __restrict__ bias,
            int n_nodes)
{
    long long idx = (long long)blockIdx.x * blockDim.x + threadIdx.x;
    long long total = (long long)n_nodes * 32;   // float4 granules
    if (idx >= total) return;
    int node = (int)(idx >> 5);
    int f    = ((int)idx & 31) << 2;
    float di = dinv[node];
    float4 gv = *(const float4*)(g   + idx * 4);
    float4 av = *(const float4*)(acc + idx * 4);
    float4 bv = *(const float4*)(bias + f);
    float4 o;
    o.x = fmaxf(fmaf(di, av.x + gv.x, bv.x), 0.0f);
    o.y = fmaxf(fmaf(di, av.y + gv.y, bv.y), 0.0f);
    o.z = fmaxf(fmaf(di, av.z + gv.z, bv.z), 0.0f);
    o.w = fmaxf(fmaf(di, av.w + gv.w, bv.w), 0.0f);
    *(float4*)(acc + idx * 4) = o;
}

// ---------------------------------------------------------------------------
// Host orchestration
// ---------------------------------------------------------------------------
extern "C" void kernel_launch(void* const* d_in, const int* in_sizes, int n_in,
                              void* d_out, int out_size, void* d_ws, size_t ws_size,
                              hipStream_t stream)
{
    const float* x  = (const float*)d_in[0];
    const int*   ei = (const int*)d_in[1];
    const float* W1 = (const float*)d_in[2];
    const float* b1 = (const float*)d_in[3];
    const float* W2 = (const float*)d_in[4];
    const float* b2 = (const float*)d_in[5];
    const float* Wc = (const float*)d_in[6];
    const float* bc = (const float*)d_in[7];
    float* out = (float*)d_out;

    const int N = in_sizes[0] / 128;   // 50000
    const int E = in_sizes[1] / 2;     // 800000

    const int* src = ei;
    const int* dst = ei + E;

    // workspace: dinv[N] | bufA[N*128] | bufB[N*128]
    float* dinv = (float*)d_ws;
    size_t npad = ((size_t)N + 255) & ~(size_t)255;
    float* bufA = dinv + npad;
    float* bufB = bufA + (size_t)N * 128;

    const int mtiles = (N + 15) / 16;
    const long long featElems = (long long)N * 32;               // float4 units
    const int gridN   = (N + 255) / 256;
    const int gridE   = (E + 255) / 256;
    const int gridE32 = (int)(((long long)E * 32 + 255) / 256);
    const int gridF   = (int)((featElems + 255) / 256);

    // degrees -> dinv = rsqrt(in_degree + 1)
    gcn_init_dinv<<<gridN, 256, 0, stream>>>(dinv, N);
    gcn_degree  <<<gridE, 256, 0, stream>>>(dst, dinv, E);
    gcn_rsqrt   <<<gridN, 256, 0, stream>>>(dinv, N);

    // ---- layer 1: g1 = (x * dinv) @ W1 ----
    gcn_gemm_wmma_f32<128, true, false><<<mtiles, 256, 0, stream>>>(
        x, W1, dinv, nullptr, bufA, N);
    (void)hipMemsetAsync(bufB, 0, (size_t)N * 128 * sizeof(float), stream);
    gcn_scatter<<<gridE32, 256, 0, stream>>>(src, dst, bufA, bufB, E);
    gcn_combine<<<gridF, 256, 0, stream>>>(bufA, bufB, dinv, b1, N);   // y1 in bufB

    // ---- layer 2: g2 = (y1 * dinv) @ W2 ----
    gcn_gemm_wmma_f32<128, true, false><<<mtiles, 256, 0, stream>>>(
        bufB, W2, dinv, nullptr, bufA, N);
    (void)hipMemsetAsync(bufB, 0, (size_t)N * 128 * sizeof(float), stream);
    gcn_scatter<<<gridE32, 256, 0, stream>>>(src, dst, bufA, bufB, E);
    gcn_combine<<<gridF, 256, 0, stream>>>(bufA, bufB, dinv, b2, N);   // y2 in bufB

    // ---- classifier head: out = y2 @ Wc + bc ----
    gcn_gemm_wmma_f32<64, false, true><<<mtiles, 256, 0, stream>>>(
        bufB, Wc, nullptr, bc, out, N);
}